// SetAbstraction_53764400611470
// MI455X (gfx1250) — compile-verified
//
#include <hip/hip_runtime.h>
#include <hip/hip_bf16.h>
#include <math.h>

// Problem constants (match reference)
#define Bb   8
#define Nn   8192
#define Mm   2048
#define Gg   32
#define CINc 64
#define COUTc 128
#define PI_F 3.14159265358979323846f

// Workspace layout (float offsets)
#define OFF_F1   0            // 8*128*8192  = 8388608
#define OFF_DP   8388608      // 8*3*2048*32 = 1572864
#define OFF_Z    9961472      // 8*2048*32   = 524288
#define OFF_S1   10485760     // sum1[128], sq1[128]
#define OFF_SS1  10486016     // scale1[128], shift1[128]
#define OFF_S2   10486272     // spe,qpe,sd,qd each [128]
#define OFF_SS2  10486784     // scpe,shpe,scd,shd each [128]

typedef __attribute__((ext_vector_type(2))) float v2f;
typedef __attribute__((ext_vector_type(8))) float v8f;

#if defined(__AMDGCN__) && __has_builtin(__builtin_amdgcn_wmma_f32_16x16x4_f32)
#define USE_WMMA 1
#else
#define USE_WMMA 0
#endif

// ---------------------------------------------------------------- zero stats
__global__ void k_zero(float* st1, float* st2) {
  int t = threadIdx.x;
  if (t < 256) st1[t] = 0.f;
  for (int i = t; i < 512; i += 256) st2[i] = 0.f;
}

// ------------------------------------------- GEMM1: f1raw = W1@f + b1, stats
// grid 256 x block 256. Wave w -> channel tile o0 = 16w. Block covers 256 cols.
__global__ __launch_bounds__(256) void k_gemm1(
    const float* __restrict__ f, const float* __restrict__ W1,
    const float* __restrict__ b1, float* __restrict__ f1,
    float* __restrict__ st1) {
  const int lane = threadIdx.x & 31, wv = threadIdx.x >> 5;
  const int hlf = lane >> 4, l16 = lane & 15;
  const int o0 = wv * 16;
  float sums[8], sqs[8];
#pragma unroll
  for (int r = 0; r < 8; ++r) { sums[r] = 0.f; sqs[r] = 0.f; }
#if USE_WMMA
  v2f afr[16];  // W1 fragments: constant across column tiles
#pragma unroll
  for (int kk = 0; kk < 16; ++kk) {
    int ka = kk * 4 + 2 * hlf;
    afr[kk].x = W1[(o0 + l16) * CINc + ka];
    afr[kk].y = W1[(o0 + l16) * CINc + ka + 1];
  }
#endif
  for (int nt = 0; nt < 16; ++nt) {
    const int gcol0 = blockIdx.x * 256 + nt * 16;
    const int bb = gcol0 / Nn;
    const int nc = gcol0 % Nn;
    float accv[8];
#if USE_WMMA
    v8f acc = {0.f, 0.f, 0.f, 0.f, 0.f, 0.f, 0.f, 0.f};
    const float* fb = f + (size_t)bb * CINc * Nn + nc + l16;
#pragma unroll
    for (int kk = 0; kk < 16; ++kk) {
      int ka = kk * 4 + 2 * hlf;
      v2f bf;
      bf.x = fb[(size_t)ka * Nn];
      bf.y = fb[(size_t)(ka + 1) * Nn];
      acc = __builtin_amdgcn_wmma_f32_16x16x4_f32(
          false, afr[kk], false, bf, (short)0, acc, false, false);
    }
#pragma unroll
    for (int r = 0; r < 8; ++r) accv[r] = acc[r];
#else
#pragma unroll
    for (int r = 0; r < 8; ++r) {
      int ch = o0 + r + hlf * 8;
      const float* wrow = W1 + ch * CINc;
      const float* fc = f + (size_t)bb * CINc * Nn + nc + l16;
      float s = 0.f;
      for (int k = 0; k < CINc; ++k) s += wrow[k] * fc[(size_t)k * Nn];
      accv[r] = s;
    }
#endif
#pragma unroll
    for (int r = 0; r < 8; ++r) {
      int ch = o0 + r + hlf * 8;
      float val = accv[r] + b1[ch];
      f1[((size_t)bb * COUTc + ch) * Nn + nc + l16] = val;
      sums[r] += val;
      sqs[r] += val * val;
    }
  }
#pragma unroll
  for (int r = 0; r < 8; ++r) {
    float s = sums[r], q = sqs[r];
    for (int off = 8; off >= 1; off >>= 1) {
      s += __shfl_xor(s, off, 32);
      q += __shfl_xor(q, off, 32);
    }
    if (l16 == 0) {
      int ch = o0 + r + hlf * 8;
      atomicAdd(&st1[ch], s);
      atomicAdd(&st1[128 + ch], q);
    }
  }
}

// ------------------------------------------------------------ finalize BN1
__global__ void k_fin1(const float* st1, const float* g1, const float* bt1,
                       float* ss1) {
  int t = threadIdx.x;
  if (t >= COUTc) return;
  const float cnt = 65536.f;  // B*N
  float mean = st1[t] / cnt;
  float var = st1[128 + t] / cnt - mean * mean;
  float inv = rsqrtf(var + 1e-5f);
  float sc = g1[t] * inv;
  ss1[t] = sc;
  ss1[128 + t] = bt1[t] - mean * sc;
}

// ------------------------------------------------- BN+ReLU in place on f1
__global__ __launch_bounds__(256) void k_bnrelu(float* f1, const float* ss1) {
  size_t i = (size_t)blockIdx.x * 256 + threadIdx.x;
  int ch = (int)((i >> 13) & 127);  // N = 8192 = 2^13
  float v = f1[i] * ss1[ch] + ss1[128 + ch];
  f1[i] = fmaxf(v, 0.f);
}

// ---------------------------------------------- ZEmb: one wave32 per (b,m)
__global__ __launch_bounds__(256) void k_zemb(
    const float* __restrict__ p, const int* __restrict__ idxp,
    const int* __restrict__ nbr, const float* __restrict__ Wz,
    const float* __restrict__ bz, float* __restrict__ dpw,
    float* __restrict__ zw) {
  __shared__ float sgrid[8][6][6][4];
  __shared__ float swz[100];
  __shared__ float sbz;
  const int tid = threadIdx.x;
  if (tid < 100) swz[tid] = Wz[tid];
  if (tid == 100) sbz = bz[0];
  for (int i = tid; i < 8 * 144; i += 256) ((float*)sgrid)[i] = 0.f;
  __syncthreads();

  const int lane = tid & 31, wv = tid >> 5;
  const int grp = blockIdx.x * 8 + wv;
  const int b = grp / Mm, mi = grp % Mm;
  const int anchor = idxp[b * Mm + mi];
  const float axp = p[((size_t)b * Nn + anchor) * 3 + 0];
  const float ayp = p[((size_t)b * Nn + anchor) * 3 + 1];
  const float azp = p[((size_t)b * Nn + anchor) * 3 + 2];
  const int nb = nbr[((size_t)b * Mm + mi) * Gg + lane];
  float px = p[((size_t)b * Nn + nb) * 3 + 0] - axp;
  float py = p[((size_t)b * Nn + nb) * 3 + 1] - ayp;
  float pz = p[((size_t)b * Nn + nb) * 3 + 2] - azp;
  const size_t mg = (size_t)mi * Gg + lane;
  dpw[((size_t)(b * 3) + 0) * ((size_t)Mm * Gg) + mg] = px;
  dpw[((size_t)(b * 3) + 1) * ((size_t)Mm * Gg) + mg] = py;
  dpw[((size_t)(b * 3) + 2) * ((size_t)Mm * Gg) + mg] = pz;

  float rho = sqrtf(px * px + py * py + pz * pz);
  // argmax over wave (first-index tiebreak)
  float best = rho; int bl = lane;
#pragma unroll
  for (int off = 16; off >= 1; off >>= 1) {
    float orr = __shfl_xor(best, off, 32);
    int ol = __shfl_xor(bl, off, 32);
    if (orr > best || (orr == best && ol < bl)) { best = orr; bl = ol; }
  }
  float dix = __shfl(px, bl, 32);
  float diy = __shfl(py, bl, 32);
  float diz = __shfl(pz, bl, 32);
  float dn = sqrtf(dix * dix + diy * diy + diz * diz) + 1e-12f;
  dix /= dn; diy /= dn; diz /= dn;
  // Rodrigues to +z
  float an = sqrtf(dix * dix + diy * diy + diz * diz) + 1e-8f;
  float a0 = dix / an, a1 = diy / an, a2 = diz / an;
  float vx = a1, vy = -a0, vz = 0.f;  // cross(a, z)
  float s = sqrtf(vx * vx + vy * vy + vz * vz);
  float c = a2;
  bool near_zero = (s < 1e-8f) && (c > 0.f);
  bool near_pi = (s < 1e-8f) && (c < 0.f);
  if (near_pi) {
    float rx, ry, rz;
    if (fabsf(a0) < 0.9f) { rx = 1.f; ry = 0.f; rz = 0.f; }
    else                  { rx = 0.f; ry = 1.f; rz = 0.f; }
    float wx = a1 * rz - a2 * ry, wy = a2 * rx - a0 * rz, wz2 = a0 * ry - a1 * rx;
    float wn = sqrtf(wx * wx + wy * wy + wz2 * wz2) + 1e-8f;
    vx = wx / wn; vy = wy / wn; vz = wz2 / wn;
  }
  float kn = sqrtf(vx * vx + vy * vy + vz * vz) + 1e-8f;
  float kx = vx / kn, ky = vy / kn, kz = vz / kn;
  float Km[3][3] = {{0.f, -kz, ky}, {kz, 0.f, -kx}, {-ky, kx, 0.f}};
  float K2[3][3], R[3][3];
#pragma unroll
  for (int i = 0; i < 3; ++i)
#pragma unroll
    for (int j = 0; j < 3; ++j)
      K2[i][j] = Km[i][0] * Km[0][j] + Km[i][1] * Km[1][j] + Km[i][2] * Km[2][j];
#pragma unroll
  for (int i = 0; i < 3; ++i)
#pragma unroll
    for (int j = 0; j < 3; ++j) {
      float id = (i == j) ? 1.f : 0.f;
      R[i][j] = near_zero ? id : (id + Km[i][j] * s + K2[i][j] * (1.f - c));
    }
  float dx = R[0][0] * px + R[0][1] * py + R[0][2] * pz;
  float dy = R[1][0] * px + R[1][1] * py + R[1][2] * pz;
  float dz = R[2][0] * px + R[2][1] * py + R[2][2] * pz;
  float r = fmaxf(sqrtf(dx * dx + dy * dy + dz * dz), 1e-12f);
  float ct = fminf(fmaxf(dz / r, -1.f + 1e-6f), 1.f - 1e-6f);
  float theta = acosf(ct);
  float phi = atan2f(dy, dx);
  phi = fmodf(phi + 2.f * PI_F, 2.f * PI_F);
  int it = min(max((int)floorf(theta / (PI_F / 6.f)), 0), 5);
  int ip = min(max((int)floorf(phi / (2.f * PI_F / 6.f)), 0), 5);
  float rmax = r;
#pragma unroll
  for (int off = 16; off >= 1; off >>= 1) rmax = fmaxf(rmax, __shfl_xor(rmax, off, 32));
  float hfac = fmaxf(0.5f - r / (2.f * (rmax + 1e-12f)), 0.f) + 0.5f;
  atomicAdd(&sgrid[wv][it][ip][0], dx);
  atomicAdd(&sgrid[wv][it][ip][1], dy);
  atomicAdd(&sgrid[wv][it][ip][2], dz);
  atomicAdd(&sgrid[wv][it][ip][3], hfac);
  __syncthreads();
  // 5x5 conv (theta zero-pad, phi circular) evaluated only at this lane's bin
  float y = 0.f;
#pragma unroll
  for (int cc = 0; cc < 4; ++cc)
#pragma unroll
    for (int kh = 0; kh < 5; ++kh) {
      int hh = it + kh - 2;
      if (hh < 0 || hh >= 6) continue;
#pragma unroll
      for (int kw = 0; kw < 5; ++kw) {
        int wc = (ip + kw + 4) % 6;
        y += sgrid[wv][hh][wc][cc] * swz[(cc * 5 + kh) * 5 + kw];
      }
    }
  zw[((size_t)b * Mm + mi) * Gg + lane] = y + sbz;
}

// ---------------------------- pe tile: W2(16x4 tile) @ dpz(4x16 cols), K=4
__device__ inline float dpz_ld(const float* dpw, const float* zw, int b, int mi,
                               int i, int g) {
  if (i < 3) return dpw[(((size_t)(b * 3) + i) * Mm + mi) * Gg + g];
  return zw[((size_t)b * Mm + mi) * Gg + g];
}

__device__ inline void pe_tile(const float* W2, const float* dpw,
                               const float* zw, int b, int mi, int g0, int o0,
                               int hlf, int l16, float accv[8]) {
#if USE_WMMA
  v2f a, bf;
  int ia = 2 * hlf;
  a.x = W2[(o0 + l16) * 4 + ia];
  a.y = W2[(o0 + l16) * 4 + ia + 1];
  bf.x = dpz_ld(dpw, zw, b, mi, ia, g0 + l16);
  bf.y = dpz_ld(dpw, zw, b, mi, ia + 1, g0 + l16);
  v8f c = {0.f, 0.f, 0.f, 0.f, 0.f, 0.f, 0.f, 0.f};
  c = __builtin_amdgcn_wmma_f32_16x16x4_f32(false, a, false, bf, (short)0, c,
                                            false, false);
#pragma unroll
  for (int r = 0; r < 8; ++r) accv[r] = c[r];
#else
  float d0 = dpz_ld(dpw, zw, b, mi, 0, g0 + l16);
  float d1 = dpz_ld(dpw, zw, b, mi, 1, g0 + l16);
  float d2 = dpz_ld(dpw, zw, b, mi, 2, g0 + l16);
  float d3 = dpz_ld(dpw, zw, b, mi, 3, g0 + l16);
#pragma unroll
  for (int r = 0; r < 8; ++r) {
    int ch = o0 + r + hlf * 8;
    accv[r] = W2[ch * 4] * d0 + W2[ch * 4 + 1] * d1 + W2[ch * 4 + 2] * d2 +
              W2[ch * 4 + 3] * d3;
  }
#endif
}

// ------------------------- pass 1 over (B,M,G): stats of pe_raw and fj - fi
__global__ __launch_bounds__(256) void k_stats2(
    const float* __restrict__ f1, const float* __restrict__ dpw,
    const float* __restrict__ zw, const int* __restrict__ idxp,
    const int* __restrict__ nbr, const float* __restrict__ W2,
    const float* __restrict__ b2, float* __restrict__ st2) {
  const int lane = threadIdx.x & 31, wv = threadIdx.x >> 5;
  const int hlf = lane >> 4, l16 = lane & 15;
  const int o0 = wv * 16;
  float spe[8], qpe[8], sd[8], qd[8];
#pragma unroll
  for (int r = 0; r < 8; ++r) { spe[r] = 0.f; qpe[r] = 0.f; sd[r] = 0.f; qd[r] = 0.f; }
  for (int j = 0; j < 32; ++j) {
    int grp = blockIdx.x * 32 + j;
    int b = grp / Mm, mi = grp % Mm;
    int anchor = idxp[b * Mm + mi];
    float av[8];
#pragma unroll
    for (int r = 0; r < 8; ++r)
      av[r] = f1[((size_t)b * COUTc + (o0 + r + hlf * 8)) * Nn + anchor];
    for (int gt = 0; gt < 2; ++gt) {
      int g = gt * 16 + l16;
      int nb = nbr[((size_t)b * Mm + mi) * Gg + g];
      float accv[8];
      pe_tile(W2, dpw, zw, b, mi, gt * 16, o0, hlf, l16, accv);
#pragma unroll
      for (int r = 0; r < 8; ++r) {
        int ch = o0 + r + hlf * 8;
        float pe = accv[r] + b2[ch];
        spe[r] += pe; qpe[r] += pe * pe;
        float d = f1[((size_t)b * COUTc + ch) * Nn + nb] - av[r];
        sd[r] += d; qd[r] += d * d;
      }
    }
  }
#pragma unroll
  for (int r = 0; r < 8; ++r) {
    float a = spe[r], bq = qpe[r], cq = sd[r], dq = qd[r];
    for (int off = 8; off >= 1; off >>= 1) {
      a += __shfl_xor(a, off, 32);
      bq += __shfl_xor(bq, off, 32);
      cq += __shfl_xor(cq, off, 32);
      dq += __shfl_xor(dq, off, 32);
    }
    if (l16 == 0) {
      int ch = o0 + r + hlf * 8;
      atomicAdd(&st2[ch], a);
      atomicAdd(&st2[128 + ch], bq);
      atomicAdd(&st2[256 + ch], cq);
      atomicAdd(&st2[384 + ch], dq);
    }
  }
}

// ------------------------------------------------------------ finalize BN2
__global__ void k_fin2(const float* st2, const float* g2, const float* bt2,
                       float* ss2) {
  int t = threadIdx.x;
  if (t >= COUTc) return;
  const float cnt = 524288.f;  // B*M*G
  float mp = st2[t] / cnt;
  float vp = st2[128 + t] / cnt - mp * mp;
  float ip = rsqrtf(vp + 1e-5f);
  float scp = g2[t] * ip;
  ss2[t] = scp;
  ss2[128 + t] = bt2[t] - mp * scp;
  float md = st2[256 + t] / cnt;
  float vd = st2[384 + t] / cnt - md * md;
  float id = rsqrtf(vd + 1e-5f);
  ss2[256 + t] = id;
  ss2[384 + t] = -md * id;
}

// --------- final: recompute pe & d, apply BNs, fout, max over G, store out
__global__ __launch_bounds__(256) void k_final(
    const float* __restrict__ f1, const float* __restrict__ dpw,
    const float* __restrict__ zw, const int* __restrict__ idxp,
    const int* __restrict__ nbr, const float* __restrict__ W2,
    const float* __restrict__ b2, const float* __restrict__ ss2,
    float* __restrict__ out) {
  const int lane = threadIdx.x & 31, wv = threadIdx.x >> 5;
  const int hlf = lane >> 4, l16 = lane & 15;
  const int o0 = wv * 16;
  const int grp = blockIdx.x;
  const int b = grp / Mm, mi = grp % Mm;
  const int anchor = idxp[b * Mm + mi];
  float av[8], scp[8], shp[8], scd[8], shd[8], bb2[8];
#pragma unroll
  for (int r = 0; r < 8; ++r) {
    int ch = o0 + r + hlf * 8;
    av[r] = f1[((size_t)b * COUTc + ch) * Nn + anchor];
    scp[r] = ss2[ch];
    shp[r] = ss2[128 + ch];
    scd[r] = ss2[256 + ch];
    shd[r] = ss2[384 + ch];
    bb2[r] = b2[ch];
  }
  float mx[8];
#pragma unroll
  for (int r = 0; r < 8; ++r) mx[r] = -3.4e38f;
  for (int gt = 0; gt < 2; ++gt) {
    int g = gt * 16 + l16;
    int nb = nbr[((size_t)b * Mm + mi) * Gg + g];
    float accv[8];
    pe_tile(W2, dpw, zw, b, mi, gt * 16, o0, hlf, l16, accv);
#pragma unroll
    for (int r = 0; r < 8; ++r) {
      int ch = o0 + r + hlf * 8;
      float pe = accv[r] + bb2[r];
      float pen = fmaxf(pe * scp[r] + shp[r], 0.f);
      float d = f1[((size_t)b * COUTc + ch) * Nn + nb] - av[r];
      mx[r] = fmaxf(mx[r], pen + d * scd[r] + shd[r]);
    }
  }
#pragma unroll
  for (int r = 0; r < 8; ++r) {
    float v = mx[r];
    for (int off = 8; off >= 1; off >>= 1) v = fmaxf(v, __shfl_xor(v, off, 32));
    if (l16 == 0) {
      int ch = o0 + r + hlf * 8;
      out[((size_t)b * COUTc + ch) * Mm + mi] = v;
    }
  }
}

// ---------------------------------------------------------------- launcher
extern "C" void kernel_launch(void* const* d_in, const int* in_sizes, int n_in,
                              void* d_out, int out_size, void* d_ws,
                              size_t ws_size, hipStream_t stream) {
  (void)in_sizes; (void)n_in; (void)out_size; (void)ws_size;
  const float* p   = (const float*)d_in[0];
  const float* f   = (const float*)d_in[1];
  const int* idx   = (const int*)d_in[2];
  const int* nbr   = (const int*)d_in[3];
  const float* W1  = (const float*)d_in[4];
  const float* b1  = (const float*)d_in[5];
  const float* g1  = (const float*)d_in[6];
  const float* bt1 = (const float*)d_in[7];
  const float* Wz  = (const float*)d_in[8];
  const float* bz  = (const float*)d_in[9];
  const float* W2  = (const float*)d_in[10];
  const float* b2  = (const float*)d_in[11];
  const float* g2  = (const float*)d_in[12];
  const float* bt2 = (const float*)d_in[13];
  float* ws = (float*)d_ws;
  float* f1  = ws + OFF_F1;
  float* dpw = ws + OFF_DP;
  float* zw  = ws + OFF_Z;
  float* st1 = ws + OFF_S1;
  float* ss1 = ws + OFF_SS1;
  float* st2 = ws + OFF_S2;
  float* ss2 = ws + OFF_SS2;
  float* out = (float*)d_out;

  k_zero<<<1, 256, 0, stream>>>(st1, st2);
  k_gemm1<<<256, 256, 0, stream>>>(f, W1, b1, f1, st1);
  k_fin1<<<1, 128, 0, stream>>>(st1, g1, bt1, ss1);
  k_bnrelu<<<32768, 256, 0, stream>>>(f1, ss1);
  k_zemb<<<2048, 256, 0, stream>>>(p, idx, nbr, Wz, bz, dpw, zw);
  k_stats2<<<512, 256, 0, stream>>>(f1, dpw, zw, idx, nbr, W2, b2, st2);
  k_fin2<<<1, 128, 0, stream>>>(st2, g2, bt2, ss2);
  k_final<<<16384, 256, 0, stream>>>(f1, dpw, zw, idx, nbr, W2, b2, ss2, out);
}